// Attention_59949153518227
// MI455X (gfx1250) — compile-verified
//
#include <hip/hip_runtime.h>

// ---------------------------------------------------------------------------
// MI455X / gfx1250, wave32. Full fp32 pipeline using V_WMMA_F32_16X16X4_F32,
// with async global->LDS (ASYNCcnt) double-buffered tile staging.
// Problem sizes: B=8, S=512, D=1024, H=16, HD=64, T=2S=1024. All tile dims
// divide evenly -> no bounds checks anywhere.
// ---------------------------------------------------------------------------

typedef float v2f __attribute__((ext_vector_type(2)));
typedef float v8f __attribute__((ext_vector_type(8)));

#define B_  8
#define S_  512
#define D_  1024
#define H_  16
#define HD_ 64
#define T_  1024

static __device__ __forceinline__ v8f wmma4(v2f a, v2f b, v8f c) {
  // D = A(16x4 f32) * B(4x16 f32) + C(16x16 f32)
  return __builtin_amdgcn_wmma_f32_16x16x4_f32(
      false, a, false, b, (short)0, c, false, false);
}

// Async copy of 16 bytes/lane from global to LDS; tracked by ASYNCcnt.
static __device__ __forceinline__ void async_b128(unsigned lds_off,
                                                  const float* gptr) {
  asm volatile("global_load_async_to_lds_b128 %0, %1, off"
               :: "v"(lds_off), "v"(gptr) : "memory");
}
static __device__ __forceinline__ void wait_async0() {
  asm volatile("s_wait_asynccnt 0" ::: "memory");
}
// Low 32 bits of the generic (flat) address of a __shared__ object are the
// wave-relative LDS byte offset (shared aperture lives in addr[63:32]).
static __device__ __forceinline__ unsigned lds_off(const void* p) {
  return (unsigned)(size_t)p;
}

static __device__ __forceinline__ float gelu_tanh(float x) {
  float x3 = x * x * x;
  return 0.5f * x * (1.0f + tanhf(0.7978845608028654f * (x + 0.044715f * x3)));
}

// ---------------------------------------------------------------------------
// Generic fp32 WMMA GEMM:  out[M,N] = epi( pre(A)[M,K] @ W[N,K]^T + bias[N] )
//   pre_mod==1 : A'[m,k] = (1+scale[b,k])*A[m,k] + shift[b,k], b = m/512
//   A row remap: arow = (m/ra)*abstride + (m%ra) + aoff  (lda == K always)
//   epi 0: bias            epi 1: gelu(bias)
//   epi 2: zres[m*N+n] + (acc+bias)*e[b*3D + 2D + n], b = m/512
//   epi 3: head scatter  out[((b*H + n/64)*T + t)*64 + n%64], b=m/T, t=m%T
// Block = 128 threads (4 waves); tile 64x64, BK=16; wave w owns rows w*16..+15.
// Double-buffered async staging: DMA of tile t+1 overlaps WMMA of tile t.
// ---------------------------------------------------------------------------
__global__ __launch_bounds__(128) void gemm_f32_wmma(
    const float* __restrict__ A, const float* __restrict__ W,
    const float* __restrict__ bias, float* __restrict__ out,
    const float* __restrict__ e, const float* __restrict__ zres,
    int M, int N, int K,
    int ra, int abstride, int aoff,
    int pre_mod, int epi_mode)
{
  __shared__ float As[2][64][16];
  __shared__ float Ws[2][64][16];

  const int tid  = threadIdx.x;
  const int wave = tid >> 5;
  const int lane = tid & 31;
  const int half = lane >> 4;   // lanes 16-31 handle K+2 / M+8 halves
  const int lr   = lane & 15;
  const int m0   = blockIdx.y * 64;
  const int n0   = blockIdx.x * 64;

  v8f acc[4];
  #pragma unroll
  for (int i = 0; i < 4; ++i)
    #pragma unroll
    for (int j = 0; j < 8; ++j) acc[i][j] = 0.0f;

  auto stage_tiles = [&](int k0, int buf) {
    // ---- A tile ----
    if (pre_mod) {
      // AdaLN modulation needs per-element math -> register path.
      for (int idx = tid; idx < 256; idx += 128) {
        int row = idx >> 2, j4 = (idx & 3) << 2;
        int gm = m0 + row;
        long arow = (long)(gm / ra) * abstride + (gm % ra) + aoff;
        float4 av = *reinterpret_cast<const float4*>(&A[arow * (long)K + k0 + j4]);
        int bb = gm / S_;
        const float* sh = e + (long)bb * (3 * D_);
        const float* sc = sh + D_;
        int kk = k0 + j4;
        av.x = (1.0f + sc[kk + 0]) * av.x + sh[kk + 0];
        av.y = (1.0f + sc[kk + 1]) * av.y + sh[kk + 1];
        av.z = (1.0f + sc[kk + 2]) * av.z + sh[kk + 2];
        av.w = (1.0f + sc[kk + 3]) * av.w + sh[kk + 3];
        *reinterpret_cast<float4*>(&As[buf][row][j4]) = av;
      }
    } else {
      for (int idx = tid; idx < 256; idx += 128) {
        int row = idx >> 2, j4 = (idx & 3) << 2;
        int gm = m0 + row;
        long arow = (long)(gm / ra) * abstride + (gm % ra) + aoff;
        async_b128(lds_off(&As[buf][row][j4]), &A[arow * (long)K + k0 + j4]);
      }
    }
    // ---- W tile (always async) ----
    for (int idx = tid; idx < 256; idx += 128) {
      int row = idx >> 2, j4 = (idx & 3) << 2;
      async_b128(lds_off(&Ws[buf][row][j4]), &W[(long)(n0 + row) * K + k0 + j4]);
    }
  };

  const int ntiles = K >> 4;
  stage_tiles(0, 0);
  wait_async0();
  __syncthreads();

  for (int t = 0; t < ntiles; ++t) {
    const int cur = t & 1;
    if (t + 1 < ntiles) stage_tiles((t + 1) << 4, cur ^ 1);

    // ---- 16 WMMAs per wave per K-tile ----
    #pragma unroll
    for (int kk = 0; kk < 16; kk += 4) {
      v2f af = *reinterpret_cast<const v2f*>(&As[cur][wave * 16 + lr][kk + 2 * half]);
      #pragma unroll
      for (int n = 0; n < 4; ++n) {
        v2f bf = *reinterpret_cast<const v2f*>(&Ws[cur][n * 16 + lr][kk + 2 * half]);
        acc[n] = wmma4(af, bf, acc[n]);
      }
    }
    wait_async0();      // next tile's DMA done (overlapped with WMMAs above)
    __syncthreads();    // all waves done reading cur + staging done
  }

  // ---- epilogue: C layout = (vgpr r, lane) -> row r+8*half, col lane%16 ----
  #pragma unroll
  for (int n = 0; n < 4; ++n) {
    int gcol = n0 + n * 16 + lr;
    float bv = bias[gcol];
    #pragma unroll
    for (int r = 0; r < 8; ++r) {
      int gm = m0 + wave * 16 + r + 8 * half;
      float v = acc[n][r] + bv;
      if (epi_mode == 1) {
        v = gelu_tanh(v);
      } else if (epi_mode == 2) {
        int bb = gm / S_;
        v = zres[(long)gm * N + gcol] + v * e[(long)bb * (3 * D_) + 2 * D_ + gcol];
      }
      if (epi_mode == 3) {
        int bb = gm / T_, t = gm % T_;
        int h = gcol >> 6, d = gcol & 63;
        out[(((long)(bb * H_ + h)) * T_ + t) * HD_ + d] = v;
      } else {
        out[(long)gm * N + gcol] = v;
      }
    }
  }
}

// ---------------------------------------------------------------------------
// RoPE (full D, positions restart per state/action half) + concat q/k/v.
// qkv_* : (B, S, 3D).  Outputs (B, 2S, D).  One thread per channel pair.
// blockIdx.y: 0=q, 1=k, 2=v (v = plain copy).
// ---------------------------------------------------------------------------
__global__ void rope_concat(const float* __restrict__ qkv_s,
                            const float* __restrict__ qkv_a,
                            float* __restrict__ qc, float* __restrict__ kc,
                            float* __restrict__ vc)
{
  int gid  = blockIdx.x * blockDim.x + threadIdx.x;   // 0 .. B*T*(D/2)-1
  int type = blockIdx.y;
  int p = gid & 511;                 // pair index 0..511
  int t = (gid >> 9) & (T_ - 1);     // 0..1023
  int b = gid >> 19;
  int s = t & (S_ - 1);              // position within half
  const float* src = (t < S_) ? qkv_s : qkv_a;
  long sbase = ((long)b * S_ + s) * (3 * D_) + type * D_ + 2 * p;
  float x0 = src[sbase], x1 = src[sbase + 1];
  float re, im;
  if (type < 2) {
    // inv = exp(-log(512) * p / 512)
    float inv = expf(-6.238324625039508f * (float)p * (1.0f / 512.0f));
    float th = (float)s * inv;
    float c = cosf(th), sn = sinf(th);
    re = x0 * c - x1 * sn;
    im = x0 * sn + x1 * c;
  } else {
    re = x0; im = x1;
  }
  float* dst = (type == 0) ? qc : (type == 1) ? kc : vc;
  long dbase = ((long)b * T_ + t) * D_ + 2 * p;
  dst[dbase] = re;
  dst[dbase + 1] = im;
}

// ---------------------------------------------------------------------------
// Flash attention, fp32 WMMA. Q/K/V in (B,H,T,HD). O written as (B,T,D) with
// col = h*64+hd (heads merged for the out-projection GEMM).
// Block = 128 threads / 4 waves; each wave owns a 16-row q strip; 64-key tiles
// staged in LDS via async copies; online softmax via 16-lane shfl_xor.
// ---------------------------------------------------------------------------
__global__ __launch_bounds__(128) void flash_attn(
    const float* __restrict__ Q, const float* __restrict__ K,
    const float* __restrict__ V, float* __restrict__ O)
{
  __shared__ float Ks[64][64];
  __shared__ float Vs[64][64];
  __shared__ float Ps[4][16][64];

  const int tid  = threadIdx.x;
  const int wave = tid >> 5;
  const int lane = tid & 31;
  const int half = lane >> 4;
  const int lr   = lane & 15;

  const int bh = blockIdx.y;
  const int b  = bh >> 4;
  const int h  = bh & 15;
  const int q0 = blockIdx.x * 64 + wave * 16;

  const float* Qb = Q + (long)bh * T_ * HD_;
  const float* Kb = K + (long)bh * T_ * HD_;
  const float* Vb = V + (long)bh * T_ * HD_;

  // Preload this wave's 16x64 Q strip in A-fragment layout.
  v2f qf[16];
  #pragma unroll
  for (int kk = 0; kk < 16; ++kk)
    qf[kk] = *reinterpret_cast<const v2f*>(
        &Qb[(long)(q0 + lr) * HD_ + kk * 4 + 2 * half]);

  float mrow[8], lrow[8];
  v8f o[4];
  #pragma unroll
  for (int r = 0; r < 8; ++r) { mrow[r] = -3.0e38f; lrow[r] = 0.0f; }
  #pragma unroll
  for (int n = 0; n < 4; ++n)
    #pragma unroll
    for (int r = 0; r < 8; ++r) o[n][r] = 0.0f;

  for (int kt = 0; kt < T_; kt += 64) {
    __syncthreads();   // previous iteration's PV reads of Ks/Vs complete
    for (int idx = tid; idx < 1024; idx += 128) {
      int row = idx >> 4, j4 = (idx & 15) << 2;
      async_b128(lds_off(&Ks[row][j4]), &Kb[(long)(kt + row) * HD_ + j4]);
      async_b128(lds_off(&Vs[row][j4]), &Vb[(long)(kt + row) * HD_ + j4]);
    }
    wait_async0();
    __syncthreads();

    // S = Q (16x64) @ K_tile^T (64x64): 64 WMMAs per wave.
    v8f s[4];
    #pragma unroll
    for (int n = 0; n < 4; ++n)
      #pragma unroll
      for (int r = 0; r < 8; ++r) s[n][r] = 0.0f;
    #pragma unroll
    for (int kk = 0; kk < 16; ++kk) {
      int kb = kk * 4 + 2 * half;
      v2f af = qf[kk];
      #pragma unroll
      for (int n = 0; n < 4; ++n) {
        v2f bf = *reinterpret_cast<const v2f*>(&Ks[n * 16 + lr][kb]);
        s[n] = wmma4(af, bf, s[n]);
      }
    }

    // Online softmax. Row r (this lane-half) spans 16 lanes of its half.
    #pragma unroll
    for (int r = 0; r < 8; ++r) {
      float mx = -3.0e38f;
      #pragma unroll
      for (int n = 0; n < 4; ++n) {
        s[n][r] *= 0.125f;                 // 1/sqrt(64)
        mx = fmaxf(mx, s[n][r]);
      }
      mx = fmaxf(mx, __shfl_xor(mx, 1));
      mx = fmaxf(mx, __shfl_xor(mx, 2));
      mx = fmaxf(mx, __shfl_xor(mx, 4));
      mx = fmaxf(mx, __shfl_xor(mx, 8));
      float mnew = fmaxf(mrow[r], mx);
      float fac = expf(mrow[r] - mnew);
      mrow[r] = mnew;
      float rs = 0.0f;
      #pragma unroll
      for (int n = 0; n < 4; ++n) {
        float p = expf(s[n][r] - mnew);
        s[n][r] = p;
        rs += p;
      }
      rs += __shfl_xor(rs, 1);
      rs += __shfl_xor(rs, 2);
      rs += __shfl_xor(rs, 4);
      rs += __shfl_xor(rs, 8);
      lrow[r] = lrow[r] * fac + rs;
      #pragma unroll
      for (int n = 0; n < 4; ++n) o[n][r] *= fac;
    }

    // Convert P from C-layout to A-layout via per-wave LDS staging.
    #pragma unroll
    for (int n = 0; n < 4; ++n)
      #pragma unroll
      for (int r = 0; r < 8; ++r)
        Ps[wave][r + 8 * half][n * 16 + lr] = s[n][r];
    __syncthreads();

    // O += P (16x64) @ V_tile (64x64): 64 WMMAs per wave.
    #pragma unroll
    for (int kk = 0; kk < 16; ++kk) {
      int kb = kk * 4 + 2 * half;
      v2f af = *reinterpret_cast<const v2f*>(&Ps[wave][lr][kb]);
      #pragma unroll
      for (int n = 0; n < 4; ++n) {
        v2f bf;
        bf.x = Vs[kb][n * 16 + lr];
        bf.y = Vs[kb + 1][n * 16 + lr];
        o[n] = wmma4(af, bf, o[n]);
      }
    }
  }

  // Normalize and scatter to (B, T, D) with heads merged.
  #pragma unroll
  for (int n = 0; n < 4; ++n)
    #pragma unroll
    for (int r = 0; r < 8; ++r) {
      int t = blockIdx.x * 64 + wave * 16 + r + 8 * half;
      O[((long)(b * T_ + t)) * D_ + h * HD_ + n * 16 + lr] = o[n][r] / lrow[r];
    }
}

// ---------------------------------------------------------------------------
// Host-side pipeline. Workspace layout (floats), with reuse (~192 MiB total):
//   R0 [0 .. 25165824)        : qkv_s|qkv_a  -> Qh|Kh|Vh -> mlp hidden
//   R1 [25165824 .. 33554432) : qcat -> O
//   R2 [33554432 .. 41943040) : kcat -> y
//   R3 [41943040 .. 50331648) : vcat
// ---------------------------------------------------------------------------
extern "C" void kernel_launch(void* const* d_in, const int* in_sizes, int n_in,
                              void* d_out, int out_size, void* d_ws, size_t ws_size,
                              hipStream_t stream) {
  const float* state_z      = (const float*)d_in[0];
  const float* action_z     = (const float*)d_in[1];
  const float* e            = (const float*)d_in[2];
  const float* qkv_state_w  = (const float*)d_in[3];
  const float* qkv_state_b  = (const float*)d_in[4];
  const float* qkv_action_w = (const float*)d_in[5];
  const float* qkv_action_b = (const float*)d_in[6];
  const float* attn_in_w    = (const float*)d_in[7];
  const float* attn_in_b    = (const float*)d_in[8];
  const float* attn_out_w   = (const float*)d_in[9];
  const float* attn_out_b   = (const float*)d_in[10];
  const float* mlp_s_w1     = (const float*)d_in[11];
  const float* mlp_s_b1     = (const float*)d_in[12];
  const float* mlp_s_w2     = (const float*)d_in[13];
  const float* mlp_s_b2     = (const float*)d_in[14];
  const float* mlp_a_w1     = (const float*)d_in[15];
  const float* mlp_a_b1     = (const float*)d_in[16];
  const float* mlp_a_w2     = (const float*)d_in[17];
  const float* mlp_a_b2     = (const float*)d_in[18];
  float* outp = (float*)d_out;

  float* wsf = (float*)d_ws;
  const size_t QKV = (size_t)B_ * S_ * 3 * D_;   // 12,582,912
  const size_t BTD = (size_t)B_ * T_ * D_;       //  8,388,608
  float* qkv_s = wsf;
  float* qkv_a = wsf + QKV;
  float* qcat  = wsf + 2 * QKV;
  float* kcat  = qcat + BTD;
  float* vcat  = kcat + BTD;
  float* Qh    = wsf;             // reuse R0 after rope consumes qkv_s/qkv_a
  float* Kh    = wsf + BTD;
  float* Vh    = wsf + 2 * BTD;
  float* Obuf  = qcat;            // reuse after Qh GEMM consumed qcat
  float* ybuf  = kcat;            // reuse after Kh GEMM consumed kcat
  float* hbuf  = wsf;             // reuse R0 after attention consumed Q/K/Vh

  const dim3 blk(128);

  // 1-2: modulated QKV projections (state, action): (4096 x 3072 x 1024)
  gemm_f32_wmma<<<dim3(3072 / 64, 4096 / 64), blk, 0, stream>>>(
      state_z, qkv_state_w, qkv_state_b, qkv_s, e, nullptr,
      4096, 3072, 1024, 4096, 0, 0, /*pre_mod=*/1, /*epi=*/0);
  gemm_f32_wmma<<<dim3(3072 / 64, 4096 / 64), blk, 0, stream>>>(
      action_z, qkv_action_w, qkv_action_b, qkv_a, e, nullptr,
      4096, 3072, 1024, 4096, 0, 0, 1, 0);

  // 3: RoPE + concat into (B, 2S, D) q/k/v
  rope_concat<<<dim3((B_ * T_ * (D_ / 2)) / 256, 3), dim3(256), 0, stream>>>(
      qkv_s, qkv_a, qcat, kcat, vcat);

  // 4-6: attention in-projections -> head-major (B,H,T,HD)
  gemm_f32_wmma<<<dim3(1024 / 64, 8192 / 64), blk, 0, stream>>>(
      qcat, attn_in_w, attn_in_b, Qh, nullptr, nullptr,
      8192, 1024, 1024, 8192, 0, 0, 0, /*epi=*/3);
  gemm_f32_wmma<<<dim3(1024 / 64, 8192 / 64), blk, 0, stream>>>(
      kcat, attn_in_w + (size_t)D_ * D_, attn_in_b + D_, Kh, nullptr, nullptr,
      8192, 1024, 1024, 8192, 0, 0, 0, 3);
  gemm_f32_wmma<<<dim3(1024 / 64, 8192 / 64), blk, 0, stream>>>(
      vcat, attn_in_w + 2 * (size_t)D_ * D_, attn_in_b + 2 * D_, Vh, nullptr,
      nullptr, 8192, 1024, 1024, 8192, 0, 0, 0, 3);

  // 7: flash attention -> O (B,T,D) heads merged
  flash_attn<<<dim3(T_ / 64, B_ * H_), blk, 0, stream>>>(Qh, Kh, Vh, Obuf);

  // 8: out projection -> y (B,T,D)
  gemm_f32_wmma<<<dim3(1024 / 64, 8192 / 64), blk, 0, stream>>>(
      Obuf, attn_out_w, attn_out_b, ybuf, nullptr, nullptr,
      8192, 1024, 1024, 8192, 0, 0, 0, 0);

  // 9-10: state MLP (y[:, :S] rows remapped), then fused residual epilogue
  gemm_f32_wmma<<<dim3(4096 / 64, 4096 / 64), blk, 0, stream>>>(
      ybuf, mlp_s_w1, mlp_s_b1, hbuf, nullptr, nullptr,
      4096, 4096, 1024, /*ra=*/512, /*abstride=*/1024, /*aoff=*/0, 0, /*epi=*/1);
  gemm_f32_wmma<<<dim3(1024 / 64, 4096 / 64), blk, 0, stream>>>(
      hbuf, mlp_s_w2, mlp_s_b2, outp, e, state_z,
      4096, 1024, 4096, 4096, 0, 0, 0, /*epi=*/2);

  // 11-12: action MLP (y[:, S:] rows), residual epilogue into second half
  gemm_f32_wmma<<<dim3(4096 / 64, 4096 / 64), blk, 0, stream>>>(
      ybuf, mlp_a_w1, mlp_a_b1, hbuf, nullptr, nullptr,
      4096, 4096, 1024, 512, 1024, /*aoff=*/512, 0, 1);
  gemm_f32_wmma<<<dim3(1024 / 64, 4096 / 64), blk, 0, stream>>>(
      hbuf, mlp_a_w2, mlp_a_b2, outp + (size_t)B_ * S_ * D_, e, action_z,
      4096, 1024, 4096, 4096, 0, 0, 0, 2);
}